// GCN_arxiv_64278480552434
// MI455X (gfx1250) — compile-verified
//
#include <hip/hip_runtime.h>
#include <hip/hip_bf16.h>

// ---------------------------------------------------------------------------
// 3-layer GCN (ogbn-arxiv) for gfx1250 / MI455X.
//  - Dense transforms: V_WMMA_F32_16X16X4_F32, W staged in LDS with an
//    interleaved pair layout so B fragments load as single ds_load_b64.
//  - Aggregation: CSR built once (u32-atomic histogram + block scan + fill),
//    then atomic-free per-node gather (one wave per destination node,
//    512B coalesced row reads, L2-resident) + global prefetch of the next
//    neighbor row to hide dependent-load latency.
// ---------------------------------------------------------------------------

typedef __attribute__((ext_vector_type(2))) float v2f;
typedef __attribute__((ext_vector_type(8))) float v8f;

#define N_NODES 169343

__device__ __forceinline__ void atomAddF(float* p, float v) {
  __hip_atomic_fetch_add(p, v, __ATOMIC_RELAXED, __HIP_MEMORY_SCOPE_AGENT);
}

// ---------------- CSR construction ------------------------------------------
__global__ void zero_int_kernel(int* p, int n) {
  int i = blockIdx.x * blockDim.x + threadIdx.x;
  if (i < n) p[i] = 0;
}

__global__ void deg_count_kernel(const int* __restrict__ dst, int* degi, int e) {
  int i = blockIdx.x * blockDim.x + threadIdx.x;
  if (i < e) atomicAdd(&degi[dst[i]], 1);
}

__global__ void dinv_kernel(const int* __restrict__ degi, float* dinv, int n) {
  int i = blockIdx.x * blockDim.x + threadIdx.x;
  if (i < n) dinv[i] = rsqrtf((float)(degi[i] + 1));  // +1 self loop
}

__global__ void block_sum_kernel(const int* __restrict__ degi, int* partials, int n) {
  __shared__ int sm[256];
  int i = blockIdx.x * 256 + threadIdx.x;
  sm[threadIdx.x] = (i < n) ? degi[i] : 0;
  __syncthreads();
  for (int s = 128; s > 0; s >>= 1) {
    if (threadIdx.x < s) sm[threadIdx.x] += sm[threadIdx.x + s];
    __syncthreads();
  }
  if (threadIdx.x == 0) partials[blockIdx.x] = sm[0];
}

// single block of 1024 threads: exclusive scan of up to 1024 partials
__global__ __launch_bounds__(1024) void scan_partials_kernel(int* partials, int nb) {
  __shared__ int sm[1024];
  int t = threadIdx.x;
  int v = (t < nb) ? partials[t] : 0;
  sm[t] = v;
  __syncthreads();
  for (int off = 1; off < 1024; off <<= 1) {
    int add = (t >= off) ? sm[t - off] : 0;
    __syncthreads();
    sm[t] += add;
    __syncthreads();
  }
  if (t < nb) partials[t] = sm[t] - v;  // exclusive
}

__global__ void rowptr_kernel(const int* __restrict__ degi, const int* __restrict__ partials,
                              int* rowptr, int* cursor, int n) {
  __shared__ int sm[256];
  int i = blockIdx.x * 256 + threadIdx.x;
  int v = (i < n) ? degi[i] : 0;
  sm[threadIdx.x] = v;
  __syncthreads();
  for (int off = 1; off < 256; off <<= 1) {
    int add = (threadIdx.x >= off) ? sm[threadIdx.x - off] : 0;
    __syncthreads();
    sm[threadIdx.x] += add;
    __syncthreads();
  }
  if (i < n) {
    int excl = partials[blockIdx.x] + sm[threadIdx.x] - v;
    rowptr[i] = excl;
    cursor[i] = excl;
  }
}

__global__ void fill_adj_kernel(const int* __restrict__ src, const int* __restrict__ dst,
                                int* cursor, int* adj, int e) {
  int i = blockIdx.x * blockDim.x + threadIdx.x;
  if (i < e) {
    int pos = atomicAdd(&cursor[dst[i]], 1);
    adj[pos] = src[i];
  }
}

// ---------------- dense transform: H[N,Fout] = X[N,128] @ W[128,Fout] -------
// 256 threads = 8 waves; each wave owns a 16-row M-tile, loops over N-tiles.
// LDS holds W as interleaved pairs so each lane's B fragment is one b64 load:
//   pair(i, h, col) = { W[4i+h][col], W[4i+2+h][col] }   at lw2[(2i+h)*fp+col]
__global__ __launch_bounds__(256) void gemm_wmma_kernel(
    const float* __restrict__ X, const float* __restrict__ W,
    float* __restrict__ H, int nrows, int fout) {
  __shared__ v2f lw2[64 * 128];

  const int ntiles = (fout + 15) >> 4;
  const int fp = ntiles << 4;  // padded column count

  for (int p = threadIdx.x; p < 64 * fp; p += 256) {
    int i = p / (2 * fp);
    int rem = p - i * 2 * fp;
    int hh = rem / fp;
    int col = rem - hh * fp;
    v2f t = {0.0f, 0.0f};
    if (col < fout) {
      t.x = W[(4 * i + hh) * fout + col];
      t.y = W[(4 * i + 2 + hh) * fout + col];
    }
    lw2[p] = t;
  }
  __syncthreads();

  const int wave = threadIdx.x >> 5;
  const int lane = threadIdx.x & 31;
  const int half = lane >> 4;
  const int lr   = lane & 15;

  const long mbase = (long)blockIdx.x * 128 + wave * 16;
  long row  = mbase + lr;
  long rowc = row < nrows ? row : (long)nrows - 1;  // clamp reads, guard stores

  // A stripe: lane<16 holds (K0,K1) of its row; lane>=16 holds (K2,K3).
  v2f a[32];
  const float* xr = X + rowc * 128 + 2 * half;
#pragma unroll
  for (int i = 0; i < 32; ++i) a[i] = *(const v2f*)(xr + 4 * i);

  for (int j = 0; j < ntiles; ++j) {
    v8f c = {};
#pragma unroll
    for (int i = 0; i < 32; ++i) {
      v2f b = lw2[(2 * i + half) * fp + j * 16 + lr];
      c = __builtin_amdgcn_wmma_f32_16x16x4_f32(
          false, a[i], false, b, (short)0, c, false, false);
    }
    const int col = j * 16 + lr;
#pragma unroll
    for (int v = 0; v < 8; ++v) {
      long r = mbase + v + 8 * half;
      if (r < nrows && col < fout) H[r * fout + col] = c[v];
    }
  }
}

// ---------------- aggregation: atomic-free per-node gather ------------------
// One wave per destination node; lane l owns float4 chunk l (f4 = fout/4).
// out[i,:] = sum_{s in adj[i]} h[s,:]*dinv[s]*dinv[i] + h[i,:]*dinv[i]^2 + bias
__global__ __launch_bounds__(256) void gather_kernel(
    const float* __restrict__ h, const int* __restrict__ rowptr,
    const int* __restrict__ degi, const int* __restrict__ adj,
    const float* __restrict__ dinv, const float* __restrict__ bias,
    float* __restrict__ out, int f4, int n, int fout) {
  long gwave = ((long)blockIdx.x * blockDim.x + threadIdx.x) >> 5;
  int lane = threadIdx.x & 31;
  if (gwave >= n) return;
  const bool active = lane < f4;
  const float di = dinv[gwave];

  float4 acc = {0.0f, 0.0f, 0.0f, 0.0f};
  if (active) {
    float4 v = ((const float4*)(h + gwave * fout))[lane];
    float w = di * di;
    acc.x = v.x * w; acc.y = v.y * w; acc.z = v.z * w; acc.w = v.w * w;
  }

  const int start = rowptr[gwave];
  const int cnt = degi[gwave];
  for (int k = 0; k < cnt; ++k) {
    int s = adj[start + k];
    if (k + 1 < cnt) {  // prefetch next neighbor row (random L2 read)
      int sn = adj[start + k + 1];
      __builtin_prefetch(h + (long)sn * fout + lane * 4, 0, 0);
    }
    float nrm = dinv[s] * di;
    if (active) {
      float4 v = ((const float4*)(h + (long)s * fout))[lane];
      acc.x += v.x * nrm; acc.y += v.y * nrm;
      acc.z += v.z * nrm; acc.w += v.w * nrm;
    }
  }

  if (active) {
    const float* bp = bias + lane * 4;
    acc.x += bp[0]; acc.y += bp[1]; acc.z += bp[2]; acc.w += bp[3];
    ((float4*)(out + gwave * fout))[lane] = acc;
  }
}

// ---------------- batch norm ------------------------------------------------
__global__ void zero_stats_kernel(float* stats) { stats[threadIdx.x] = 0.0f; }

__global__ void bn_stats_kernel(const float* __restrict__ a, float* stats,
                                int f, int n, int rows_per_block) {
  int c = threadIdx.x;
  long r0 = (long)blockIdx.x * rows_per_block;
  float s = 0.0f, s2 = 0.0f;
  for (int k = 0; k < rows_per_block; ++k) {
    long r = r0 + k;
    if (r < n) {
      float v = a[r * f + c];
      s += v; s2 += v * v;
    }
  }
  atomAddF(&stats[c], s);
  atomAddF(&stats[f + c], s2);
}

__global__ void bn_apply_relu_kernel(const float* __restrict__ a,
                                     const float* __restrict__ stats,
                                     const float* __restrict__ gamma,
                                     const float* __restrict__ beta,
                                     float* __restrict__ out,
                                     int f, long total, float inv_n) {
  long idx = (long)blockIdx.x * blockDim.x + threadIdx.x;
  if (idx >= total) return;
  int c = (int)(idx % f);
  float mean = stats[c] * inv_n;
  float var  = stats[f + c] * inv_n - mean * mean;
  float v = (a[idx] - mean) * rsqrtf(var + 1e-5f) * gamma[c] + beta[c];
  out[idx] = v > 0.0f ? v : 0.0f;
}

// ---------------------------------------------------------------------------
extern "C" void kernel_launch(void* const* d_in, const int* in_sizes, int n_in,
                              void* d_out, int out_size, void* d_ws, size_t ws_size,
                              hipStream_t stream) {
  const int N = N_NODES;
  const int E = in_sizes[1] / 2;

  const float* x    = (const float*)d_in[0];
  const int*   esrc = (const int*)d_in[1];
  const int*   edst = esrc + E;
  const float* W1 = (const float*)d_in[2];
  const float* b1 = (const float*)d_in[3];
  const float* W2 = (const float*)d_in[4];
  const float* b2 = (const float*)d_in[5];
  const float* W3 = (const float*)d_in[6];
  const float* b3 = (const float*)d_in[7];
  const float* g1 = (const float*)d_in[8];
  const float* be1 = (const float*)d_in[9];
  const float* g2 = (const float*)d_in[10];
  const float* be2 = (const float*)d_in[11];
  float* out = (float*)d_out;

  // workspace layout
  const size_t Na = ((size_t)N + 255) & ~(size_t)255;
  const size_t Ea = ((size_t)E + 255) & ~(size_t)255;
  char* w = (char*)d_ws;
  int* degi     = (int*)w;            w += Na * 4;
  int* rowptr   = (int*)w;            w += Na * 4;
  int* cursor   = (int*)w;            w += Na * 4;
  int* adj      = (int*)w;            w += Ea * 4;
  int* partials = (int*)w;            w += 1024 * 4;
  float* dinv   = (float*)w;          w += Na * 4;
  float* stats  = (float*)w;          w += 256 * 4;
  float* buf0   = (float*)w;          w += (size_t)N * 128 * 4;
  float* buf1   = (float*)w;

  const int T = 256;
  const long NF = (long)N * 128;
  const int nb = (N + 255) / 256;  // 662 <= 1024
  #define CDIV(a, b) (int)(((long)(a) + (b) - 1) / (b))

  // ---- CSR build + normalization (once)
  zero_int_kernel<<<CDIV(N, T), T, 0, stream>>>(degi, N);
  deg_count_kernel<<<CDIV(E, T), T, 0, stream>>>(edst, degi, E);
  dinv_kernel<<<CDIV(N, T), T, 0, stream>>>(degi, dinv, N);
  block_sum_kernel<<<nb, 256, 0, stream>>>(degi, partials, N);
  scan_partials_kernel<<<1, 1024, 0, stream>>>(partials, nb);
  rowptr_kernel<<<nb, 256, 0, stream>>>(degi, partials, rowptr, cursor, N);
  fill_adj_kernel<<<CDIV(E, T), T, 0, stream>>>(esrc, edst, cursor, adj, E);

  const int gatherGrid = CDIV((long)N * 32, T);  // one wave per node

  // ---- layer 1: h1 = x@W1 -> buf0 ; gather -> buf1 ; bn+relu -> buf0
  gemm_wmma_kernel<<<CDIV(N, 128), T, 0, stream>>>(x, W1, buf0, N, 128);
  gather_kernel<<<gatherGrid, T, 0, stream>>>(buf0, rowptr, degi, adj, dinv, b1, buf1, 32, N, 128);
  zero_stats_kernel<<<1, 256, 0, stream>>>(stats);
  bn_stats_kernel<<<CDIV(N, 256), 128, 0, stream>>>(buf1, stats, 128, N, 256);
  bn_apply_relu_kernel<<<CDIV(NF, T), T, 0, stream>>>(buf1, stats, g1, be1, buf0, 128, NF, 1.0f / N);

  // ---- layer 2: h2 = buf0@W2 -> buf1 ; gather -> buf0 ; bn+relu -> buf1
  gemm_wmma_kernel<<<CDIV(N, 128), T, 0, stream>>>(buf0, W2, buf1, N, 128);
  gather_kernel<<<gatherGrid, T, 0, stream>>>(buf1, rowptr, degi, adj, dinv, b2, buf0, 32, N, 128);
  zero_stats_kernel<<<1, 256, 0, stream>>>(stats);
  bn_stats_kernel<<<CDIV(N, 256), 128, 0, stream>>>(buf0, stats, 128, N, 256);
  bn_apply_relu_kernel<<<CDIV(NF, T), T, 0, stream>>>(buf0, stats, g2, be2, buf1, 128, NF, 1.0f / N);

  // ---- layer 3: h3 = buf1@W3 -> buf0 (N x 40) ; gather -> d_out
  gemm_wmma_kernel<<<CDIV(N, 128), T, 0, stream>>>(buf1, W3, buf0, N, 40);
  gather_kernel<<<gatherGrid, T, 0, stream>>>(buf0, rowptr, degi, adj, dinv, b3, out, 10, N, 40);

  #undef CDIV
}